// Upsample_4904852652748
// MI455X (gfx1250) — compile-verified
//
#include <hip/hip_runtime.h>
#include <stdint.h>

// -------------------------------------------------------------------------
// 2x FIR upsample (upfirdn2d, up=2, k=(1,3,3,1) outer, gain=4), f32.
// Memory-bound: 128MB in + 512MB out @ 23.3TB/s => ~27us floor.
// Strategy: async-copy input tile (+halo) to LDS via CDNA5
// GLOBAL_LOAD_ASYNC_TO_LDS (ASYNCcnt path), compute the 2x2-phase FIR in
// registers, stream results out with nontemporal B128 stores.
// -------------------------------------------------------------------------

typedef float v4f __attribute__((ext_vector_type(4)));

#define AS1 __attribute__((address_space(1)))
#define AS3 __attribute__((address_space(3)))

#define H_IN   128
#define W_IN   128
#define TILE_H 16
#define TILE_W 64
#define HALO_H (TILE_H + 2)
#define HALO_W (TILE_W + 2)
#define LDS_W  68          // padded row stride (floats)

__device__ __forceinline__ void async_g2l_b32(const float* g, float* l) {
#if __has_builtin(__builtin_amdgcn_global_load_async_to_lds_b32)
    // Signature (probe-confirmed): (AS1 int* gaddr, AS3 int* lds, imm off, imm cpol)
    __builtin_amdgcn_global_load_async_to_lds_b32(
        (AS1 int*)(uintptr_t)g, (AS3 int*)l, /*offset=*/0, /*cpol=*/0);
#else
    // GV mode: vdst = LDS byte offset, vaddr = 64-bit global address
    uint32_t loff = (uint32_t)(uintptr_t)(AS3 float*)l;
    asm volatile("global_load_async_to_lds_b32 %0, %1, off"
                 :: "v"(loff), "v"(g) : "memory");
#endif
}

__device__ __forceinline__ void wait_asynccnt0() {
#if __has_builtin(__builtin_amdgcn_s_wait_asynccnt)
    __builtin_amdgcn_s_wait_asynccnt(0);
#else
    asm volatile("s_wait_asynccnt 0x0" ::: "memory");
#endif
}

__global__ __launch_bounds__(256) void upsample2x_fir_kernel(
    const float* __restrict__ x,     // (N, 128, 128) with N = B*C
    const float* __restrict__ kw,    // (4,4) FIR kernel, sums to 4
    float* __restrict__ out)         // (N, 256, 256)
{
    __shared__ float tile[HALO_H * LDS_W];

    const int img = blockIdx.z;
    const int y0  = blockIdx.y * TILE_H;
    const int x0  = blockIdx.x * TILE_W;
    const int tx  = threadIdx.x;              // 0..15  -> 4-wide input col group
    const int ty  = threadIdx.y;              // 0..15  -> input row
    const int t   = ty * 16 + tx;             // 0..255

    const float* src = x + (size_t)img * (H_IN * W_IN);

    // Flipped kernel taps (true convolution); uniform scalar loads.
    // kf[a][b] = flip(k)[a][b] = kw[(3-a)*4 + (3-b)]
#define KF(a, b) kw[(3 - (a)) * 4 + (3 - (b))]
    const float k00 = KF(0,0), k01 = KF(0,1), k02 = KF(0,2), k03 = KF(0,3);
    const float k10 = KF(1,0), k11 = KF(1,1), k12 = KF(1,2), k13 = KF(1,3);
    const float k20 = KF(2,0), k21 = KF(2,1), k22 = KF(2,2), k23 = KF(2,3);
    const float k30 = KF(3,0), k31 = KF(3,1), k32 = KF(3,2), k33 = KF(3,3);
#undef KF

    // 1) Zero-fill LDS tile (implements the image-border zero padding).
    for (int e = t; e < HALO_H * LDS_W; e += 256)
        tile[e] = 0.0f;
    __syncthreads();

    // 2) Async-copy the in-bounds halo region global -> LDS (ASYNCcnt path).
    //    rows y0-1 .. y0+TILE_H, cols x0-1 .. x0+TILE_W
    for (int e = t; e < HALO_H * HALO_W; e += 256) {
        const int r  = e / HALO_W;
        const int c  = e - r * HALO_W;
        const int gy = y0 - 1 + r;
        const int gx = x0 - 1 + c;
        if (gy >= 0 && gy < H_IN && gx >= 0 && gx < W_IN)
            async_g2l_b32(src + gy * W_IN + gx, &tile[r * LDS_W + c]);
    }
    wait_asynccnt0();
    __syncthreads();

    // 3) Compute a 2(row) x 8(col) output block per thread.
    //    Output phases: oy=2i+a, ox=2j+b use taps
    //      a=0: rows {i-1 (k row 0), i (k row 2)}   a=1: rows {i (k row 1), i+1 (k row 3)}
    //      b=0: cols {j-1, j}                       b=1: cols {j, j+1}
    const int lr = ty + 1;                 // LDS row of input row i = y0+ty
    const int lc = tx * 4 + 1;             // LDS col of input col j = x0+tx*4
    const float* rowm = &tile[(lr - 1) * LDS_W + (lc - 1)];
    const float* row0 = &tile[(lr    ) * LDS_W + (lc - 1)];
    const float* rowp = &tile[(lr + 1) * LDS_W + (lc - 1)];

    float rm[6], r0[6], rp[6];             // cols j-1 .. j+4
#pragma unroll
    for (int q = 0; q < 6; ++q) {
        rm[q] = rowm[q];
        r0[q] = row0[q];
        rp[q] = rowp[q];
    }

    float ev[8], od[8];
#pragma unroll
    for (int q = 0; q < 4; ++q) {          // output cols 2(j+q), 2(j+q)+1
        ev[2*q]   = k00*rm[q]   + k02*rm[q+1] + k20*r0[q]   + k22*r0[q+1];
        ev[2*q+1] = k01*rm[q+1] + k03*rm[q+2] + k21*r0[q+1] + k23*r0[q+2];
        od[2*q]   = k10*r0[q]   + k12*r0[q+1] + k30*rp[q]   + k32*rp[q+1];
        od[2*q+1] = k11*r0[q+1] + k13*r0[q+2] + k31*rp[q+1] + k33*rp[q+2];
    }

    // 4) Streaming (nontemporal) B128 stores: output is write-once 512MB.
    const int oy = 2 * (y0 + ty);
    const int ox = 2 * (x0 + tx * 4);
    float* dst = out + (size_t)img * (2 * H_IN * 2 * W_IN)
                     + (size_t)oy * (2 * W_IN) + ox;

    v4f e0 = { ev[0], ev[1], ev[2], ev[3] };
    v4f e1 = { ev[4], ev[5], ev[6], ev[7] };
    v4f o0 = { od[0], od[1], od[2], od[3] };
    v4f o1 = { od[4], od[5], od[6], od[7] };
    __builtin_nontemporal_store(e0, (v4f*)(dst));
    __builtin_nontemporal_store(e1, (v4f*)(dst + 4));
    __builtin_nontemporal_store(o0, (v4f*)(dst + 2 * W_IN));
    __builtin_nontemporal_store(o1, (v4f*)(dst + 2 * W_IN + 4));
}

extern "C" void kernel_launch(void* const* d_in, const int* in_sizes, int n_in,
                              void* d_out, int out_size, void* d_ws, size_t ws_size,
                              hipStream_t stream) {
    const float* x  = (const float*)d_in[0];   // (16,128,128,128) f32
    const float* kw = (const float*)d_in[1];   // (4,4) f32
    float* out      = (float*)d_out;           // (16,128,256,256) f32

    const int N = in_sizes[0] / (H_IN * W_IN); // B*C = 2048 images
    dim3 grid(W_IN / TILE_W, H_IN / TILE_H, N);   // (2, 8, 2048)
    dim3 block(16, 16, 1);                        // 256 threads = 8 wave32
    upsample2x_fir_kernel<<<grid, block, 0, stream>>>(x, kw, out);
}